// AdaptiveAttention_15968688406650
// MI455X (gfx1250) — compile-verified
//
#include <hip/hip_runtime.h>

// ---------------- problem constants ----------------
#define Bc    2
#define Sc    2048
#define HIDc  1024
#define NHc   8
#define Dc    128
#define MTOT  (Bc * Sc)                      // 4096 rows
#define OUT_ELEMS ((size_t)Bc * Sc * HIDc)   // 4194304
#define BH    (Bc * NHc)                     // 16

typedef float v2f __attribute__((ext_vector_type(2)));
typedef float v8f __attribute__((ext_vector_type(8)));
typedef int   v4i __attribute__((ext_vector_type(4)));

// ---------------- CDNA5 feature probes ----------------
#if defined(__has_builtin)
#  if __has_builtin(__builtin_amdgcn_global_load_async_to_lds_b128)
#    define HAVE_ASYNC_LDS 1
#  else
#    define HAVE_ASYNC_LDS 0
#  endif
#  if __has_builtin(__builtin_amdgcn_s_wait_asynccnt)
#    define WAIT_ASYNC(n) __builtin_amdgcn_s_wait_asynccnt(n)
#  else
#    define WAIT_ASYNC(n) asm volatile("s_wait_asynccnt " #n ::: "memory")
#  endif
#else
#  define HAVE_ASYNC_LDS 0
#  define WAIT_ASYNC(n)
#endif

static __device__ __forceinline__ void cp_async16(const void* g, void* l) {
#if HAVE_ASYNC_LDS
  // GLOBAL_LOAD_ASYNC_TO_LDS_B128: 16 bytes per lane, tracked by ASYNCcnt.
  // Builtin signature (from hipcc diagnostic): (v4i AS1*, v4i AS3*, imm, imm)
  __builtin_amdgcn_global_load_async_to_lds_b128(
      (__attribute__((address_space(1))) v4i*)g,
      (__attribute__((address_space(3))) v4i*)l, 0, 0);
#else
  (void)g; (void)l;
#endif
}

static __device__ __forceinline__ v8f wmma4(v2f a, v2f b, v8f c) {
  // D = A(16x4 f32) * B(4x16 f32) + C(16x16 f32)
  return __builtin_amdgcn_wmma_f32_16x16x4_f32(
      /*neg_a=*/false, a, /*neg_b=*/false, b,
      /*c_mod=*/(short)0, c, /*reuse_a=*/false, /*reuse_b=*/false);
}

// LDS row strides: multiple of 4 floats (16B-aligned b128 chunks), gcd(.,64)==4
// so 16 distinct rows map to 16 distinct bank groups on fragment reads.
#define STRIDE32 44    // for 32-wide K tiles
#define STRIDE128 132  // for 128-wide tiles (scores)

// =====================================================================
// Kernel 1: C[m,n] = A[m,:] . W[n,:] + bias[n] (+ residual[m,n])
// Row-major tiles for BOTH operands: f32 WMMA fragments are K-pairs, and
// B[k][n] = W[n][k] means the B-fragment is Ws[n][ka..ka+1] (contiguous).
// Double-buffered async-to-LDS pipeline when available.
// =====================================================================
__global__ __launch_bounds__(128) void gemm_xwT_kernel(
    const float* __restrict__ A, const float* __restrict__ W,
    const float* __restrict__ bias, const float* __restrict__ resid,
    float* __restrict__ Cout)
{
  __shared__ float As[2][64][STRIDE32];
  __shared__ float Ws[2][64][STRIDE32];

  const int m0   = blockIdx.x * 64;
  const int n0   = blockIdx.y * 64;
  const int tid  = threadIdx.x;
  const int lane = tid & 31;
  const int wv   = tid >> 5;       // 0..3
  const int half = lane >> 4;      // 0/1
  const int l16  = lane & 15;

  v8f acc[4];
  #pragma unroll
  for (int j = 0; j < 4; ++j) acc[j] = (v8f)0.0f;

#if HAVE_ASYNC_LDS
  // prologue: tile 0 (8 async b128 per thread per tile)
  #pragma unroll
  for (int f = tid; f < 512; f += 128) {
    const int r = f >> 3, c4 = (f & 7) << 2;
    cp_async16(A + (size_t)(m0 + r) * HIDc + c4, &As[0][r][c4]);
    cp_async16(W + (size_t)(n0 + r) * HIDc + c4, &Ws[0][r][c4]);
  }
#endif

  int buf = 0;
  for (int kc = 0; kc < HIDc; kc += 32, buf ^= 1) {
#if HAVE_ASYNC_LDS
    const bool has_next = (kc + 32) < HIDc;
    if (has_next) {
      #pragma unroll
      for (int f = tid; f < 512; f += 128) {
        const int r = f >> 3, c4 = (f & 7) << 2;
        cp_async16(A + (size_t)(m0 + r) * HIDc + kc + 32 + c4, &As[buf ^ 1][r][c4]);
        cp_async16(W + (size_t)(n0 + r) * HIDc + kc + 32 + c4, &Ws[buf ^ 1][r][c4]);
      }
      WAIT_ASYNC(8);   // 8 newest (next tile) may stay in flight; current tile done
    } else {
      WAIT_ASYNC(0);
    }
    __syncthreads();
#else
    #pragma unroll
    for (int f = tid; f < 512; f += 128) {
      const int r = f >> 3, c4 = (f & 7) << 2;
      *(float4*)&As[buf][r][c4] = *(const float4*)(A + (size_t)(m0 + r) * HIDc + kc + c4);
      *(float4*)&Ws[buf][r][c4] = *(const float4*)(W + (size_t)(n0 + r) * HIDc + kc + c4);
    }
    if (kc + 32 < HIDc) {  // pull next tiles toward L2 (global_prefetch_b8)
      __builtin_prefetch(A + (size_t)(m0 + (tid >> 1)) * HIDc + kc + 32, 0, 1);
      __builtin_prefetch(W + (size_t)(n0 + (tid >> 1)) * HIDc + kc + 32, 0, 1);
    }
    __syncthreads();
#endif

    const int arow = (wv << 4) + l16;
    #pragma unroll
    for (int kk = 0; kk < 32; kk += 4) {
      const int ka = kk + (half << 1);        // A layout: v0:K0/K2, v1:K1/K3
      const v2f a = *(const v2f*)&As[buf][arow][ka];
      #pragma unroll
      for (int j = 0; j < 4; ++j) {
        const v2f b = *(const v2f*)&Ws[buf][(j << 4) + l16][ka];
        acc[j] = wmma4(a, b, acc[j]);
      }
    }
    __syncthreads();   // all reads of this buffer done before it is refilled
  }

  const bool hasRes = (resid != nullptr);
  #pragma unroll
  for (int j = 0; j < 4; ++j) {
    const int gn = n0 + (j << 4) + l16;
    const float bv = bias[gn];
    #pragma unroll
    for (int i = 0; i < 8; ++i) {
      const int gm = m0 + (wv << 4) + (half << 3) + i;   // C layout: vgpr i -> M=i / M=8+i
      float v = acc[j][i] + bv;
      if (hasRes) v += resid[(size_t)gm * HIDc + gn];
      Cout[(size_t)gm * HIDc + gn] = v;
    }
  }
}

// =====================================================================
// Kernel 2: per-(b,h,s) L2 norms of Q and K head slices (wave per row)
// =====================================================================
__global__ __launch_bounds__(128) void head_norms_kernel(
    const float* __restrict__ Q, const float* __restrict__ K,
    float* __restrict__ qn, float* __restrict__ kn)
{
  const int row  = blockIdx.x * 4 + (threadIdx.x >> 5);  // (b*NH+h)*S + s
  const int lane = threadIdx.x & 31;
  const int s  = row & (Sc - 1);
  const int bh = row >> 11;            // /Sc
  const int b  = bh >> 3, h = bh & 7;
  const size_t base = ((size_t)(b * Sc + s)) * HIDc + h * Dc + lane * 4;
  const float4 q4 = *(const float4*)(Q + base);
  const float4 k4 = *(const float4*)(K + base);
  float qs = q4.x*q4.x + q4.y*q4.y + q4.z*q4.z + q4.w*q4.w;
  float ks = k4.x*k4.x + k4.y*k4.y + k4.z*k4.z + k4.w*k4.w;
  #pragma unroll
  for (int o = 16; o; o >>= 1) { qs += __shfl_xor(qs, o); ks += __shfl_xor(ks, o); }
  if (lane == 0) { qn[row] = sqrtf(qs); kn[row] = sqrtf(ks); }
}

// =====================================================================
// Kernel 3: raw mixed scores -> attn buffer (pre-softmax).
// Per (b,h): 64x64 tile of Q.K^T via WMMA, fused cos/euclid epilogue.
// Both tiles row-major; B-fragment = Ks[n][ka..ka+1] (contiguous b64).
// =====================================================================
__global__ __launch_bounds__(128) void scores_kernel(
    const float* __restrict__ Q, const float* __restrict__ K,
    const float* __restrict__ qn, const float* __restrict__ kn,
    const float* __restrict__ sw, float* __restrict__ attn)
{
  __shared__ float Qs[64][STRIDE128];
  __shared__ float Ks[64][STRIDE128];

  const int q0 = blockIdx.x * 64, k0 = blockIdx.y * 64;
  const int bh = blockIdx.z;
  const int b = bh >> 3, h = bh & 7;
  const int tid = threadIdx.x, lane = tid & 31;
  const int wv = tid >> 5, half = lane >> 4, l16 = lane & 15;

#if HAVE_ASYNC_LDS
  for (int f = tid; f < 2048; f += 128) {
    const int r = f >> 5, c4 = (f & 31) << 2;
    cp_async16(Q + ((size_t)(b * Sc + q0 + r)) * HIDc + h * Dc + c4, &Qs[r][c4]);
    cp_async16(K + ((size_t)(b * Sc + k0 + r)) * HIDc + h * Dc + c4, &Ks[r][c4]);
  }
  WAIT_ASYNC(0);
#else
  for (int f = tid; f < 2048; f += 128) {
    const int r = f >> 5, c4 = (f & 31) << 2;
    *(float4*)&Qs[r][c4] = *(const float4*)(Q + ((size_t)(b * Sc + q0 + r)) * HIDc + h * Dc + c4);
    *(float4*)&Ks[r][c4] = *(const float4*)(K + ((size_t)(b * Sc + k0 + r)) * HIDc + h * Dc + c4);
  }
#endif
  __syncthreads();

  v8f acc[4];
  #pragma unroll
  for (int j = 0; j < 4; ++j) acc[j] = (v8f)0.0f;

  const int arow = (wv << 4) + l16;
  #pragma unroll 4
  for (int kd = 0; kd < Dc; kd += 4) {
    const int ka = kd + (half << 1);
    const v2f a = *(const v2f*)&Qs[arow][ka];
    #pragma unroll
    for (int j = 0; j < 4; ++j) {
      const v2f bf = *(const v2f*)&Ks[(j << 4) + l16][ka];
      acc[j] = wmma4(a, bf, acc[j]);
    }
  }

  // softmax over the 3 scoring weights (tiny, recomputed per thread)
  float w0 = sw[0], w1 = sw[1], w2 = sw[2];
  const float mw = fmaxf(w0, fmaxf(w1, w2));
  const float e0 = __expf(w0 - mw), e1 = __expf(w1 - mw), e2 = __expf(w2 - mw);
  const float inv = 1.0f / (e0 + e1 + e2);
  w0 = e0 * inv; w1 = e1 * inv; w2 = e2 * inv;

  #pragma unroll
  for (int j = 0; j < 4; ++j) {
    const int gk = k0 + (j << 4) + l16;
    const float kv = kn[bh * Sc + gk];
    #pragma unroll
    for (int i = 0; i < 8; ++i) {
      const int gq = q0 + (wv << 4) + (half << 3) + i;
      const float qv = qn[bh * Sc + gq];
      const float d  = acc[j][i];
      const float cosv = d / fmaxf(qv * kv, 1e-8f);
      const float d2   = qv * qv + kv * kv - 2.0f * d;
      const float euc  = -sqrtf(fmaxf(d2, 1e-12f));
      attn[((size_t)bh * Sc + gq) * Sc + gk] = w0 * d + w1 * cosv + w2 * euc;
    }
  }
}

// =====================================================================
// Kernel 4: in-place row softmax over attn [BH*S rows of S]
// =====================================================================
__global__ __launch_bounds__(256) void softmax_rows_kernel(float* __restrict__ attn)
{
  __shared__ float red[256];
  float* p = attn + (size_t)blockIdx.x * Sc;
  const int t = threadIdx.x;
  float v[8];
  float m = -3.4e38f;
  #pragma unroll
  for (int i = 0; i < 8; ++i) { v[i] = p[t + (i << 8)]; m = fmaxf(m, v[i]); }
  red[t] = m; __syncthreads();
  for (int s = 128; s; s >>= 1) { if (t < s) red[t] = fmaxf(red[t], red[t + s]); __syncthreads(); }
  m = red[0]; __syncthreads();
  float sum = 0.0f;
  #pragma unroll
  for (int i = 0; i < 8; ++i) { v[i] = __expf(v[i] - m); sum += v[i]; }
  red[t] = sum; __syncthreads();
  for (int s = 128; s; s >>= 1) { if (t < s) red[t] += red[t + s]; __syncthreads(); }
  const float inv = 1.0f / red[0];
  #pragma unroll
  for (int i = 0; i < 8; ++i) p[t + (i << 8)] = v[i] * inv;
}

// =====================================================================
// Kernel 5: ctx = attn @ V_head, per (b,h); K-dim = S chunked by 32.
// attn tile row-major (async-copyable); V tile transposed on STORE so the
// B-fragment read Vt[n][ka..ka+1] is a contiguous aligned b64.
// =====================================================================
__global__ __launch_bounds__(128) void ctx_kernel(
    const float* __restrict__ attn, const float* __restrict__ V,
    float* __restrict__ CTX)
{
  __shared__ float Ac[64][STRIDE32];
  __shared__ float Vt[64][STRIDE32];   // Vt[n][k]

  const int q0 = blockIdx.x * 64;
  const int n0 = blockIdx.y * 64;          // 0 or 64 within D=128
  const int bh = blockIdx.z;
  const int b = bh >> 3, h = bh & 7;
  const int tid = threadIdx.x, lane = tid & 31;
  const int wv = tid >> 5, half = lane >> 4, l16 = lane & 15;

  v8f acc[4];
  #pragma unroll
  for (int j = 0; j < 4; ++j) acc[j] = (v8f)0.0f;

  for (int kc = 0; kc < Sc; kc += 32) {
#if HAVE_ASYNC_LDS
    #pragma unroll
    for (int f = tid; f < 512; f += 128) {   // attn tile 64x32
      const int r = f >> 3, c4 = (f & 7) << 2;
      cp_async16(attn + ((size_t)bh * Sc + q0 + r) * Sc + kc + c4, &Ac[r][c4]);
    }
#else
    #pragma unroll
    for (int f = tid; f < 512; f += 128) {
      const int r = f >> 3, c4 = (f & 7) << 2;
      *(float4*)&Ac[r][c4] =
          *(const float4*)(attn + ((size_t)bh * Sc + q0 + r) * Sc + kc + c4);
    }
#endif
    #pragma unroll
    for (int f = tid; f < 512; f += 128) {   // V tile 32x64, transposed store
      const int r = f >> 4, c4 = (f & 15) << 2;
      const float4 v = *(const float4*)(V + ((size_t)(b * Sc + kc + r)) * HIDc + h * Dc + n0 + c4);
      Vt[c4 + 0][r] = v.x; Vt[c4 + 1][r] = v.y;
      Vt[c4 + 2][r] = v.z; Vt[c4 + 3][r] = v.w;
    }
#if HAVE_ASYNC_LDS
    WAIT_ASYNC(0);
#endif
    __syncthreads();

    const int arow = (wv << 4) + l16;
    #pragma unroll
    for (int kk = 0; kk < 32; kk += 4) {
      const int ka = kk + (half << 1);
      const v2f a = *(const v2f*)&Ac[arow][ka];
      #pragma unroll
      for (int j = 0; j < 4; ++j) {
        const v2f bf = *(const v2f*)&Vt[(j << 4) + l16][ka];
        acc[j] = wmma4(a, bf, acc[j]);
      }
    }
    __syncthreads();
  }

  #pragma unroll
  for (int j = 0; j < 4; ++j) {
    const int gn = h * Dc + n0 + (j << 4) + l16;
    #pragma unroll
    for (int i = 0; i < 8; ++i) {
      const int gq = q0 + (wv << 4) + (half << 3) + i;
      CTX[(size_t)(b * Sc + gq) * HIDc + gn] = acc[j][i];
    }
  }
}

// =====================================================================
// Kernel 6: in-place LayerNorm over 1024-wide rows of d_out
// =====================================================================
__global__ __launch_bounds__(256) void layernorm_kernel(
    float* __restrict__ out, const float* __restrict__ gamma,
    const float* __restrict__ beta)
{
  __shared__ float red[256];
  float* p = out + (size_t)blockIdx.x * HIDc;
  const int t = threadIdx.x;
  float v[4]; float s = 0.0f;
  #pragma unroll
  for (int i = 0; i < 4; ++i) { v[i] = p[t + (i << 8)]; s += v[i]; }
  red[t] = s; __syncthreads();
  for (int k = 128; k; k >>= 1) { if (t < k) red[t] += red[t + k]; __syncthreads(); }
  const float mu = red[0] * (1.0f / HIDc); __syncthreads();
  float s2 = 0.0f;
  #pragma unroll
  for (int i = 0; i < 4; ++i) { const float d = v[i] - mu; s2 += d * d; }
  red[t] = s2; __syncthreads();
  for (int k = 128; k; k >>= 1) { if (t < k) red[t] += red[t + k]; __syncthreads(); }
  const float rstd = rsqrtf(red[0] * (1.0f / HIDc) + 1e-5f);
  #pragma unroll
  for (int i = 0; i < 4; ++i) {
    const int c = t + (i << 8);
    p[c] = (v[i] - mu) * rstd * gamma[c] + beta[c];
  }
}

// =====================================================================
extern "C" void kernel_launch(void* const* d_in, const int* in_sizes, int n_in,
                              void* d_out, int out_size, void* d_ws, size_t ws_size,
                              hipStream_t stream)
{
  const float* query  = (const float*)d_in[0];
  const float* key_in = (const float*)d_in[1];
  const float* value  = (const float*)d_in[2];
  const float* Wq = (const float*)d_in[3];  const float* bq = (const float*)d_in[4];
  const float* Wk = (const float*)d_in[5];  const float* bk = (const float*)d_in[6];
  const float* Wv = (const float*)d_in[7];  const float* bv = (const float*)d_in[8];
  const float* Wo = (const float*)d_in[9];  const float* bo = (const float*)d_in[10];
  const float* sw    = (const float*)d_in[11];
  const float* gamma = (const float*)d_in[12];
  const float* beta  = (const float*)d_in[13];

  float* ws  = (float*)d_ws;
  float* Q   = ws;
  float* K   = Q + OUT_ELEMS;
  float* V   = K + OUT_ELEMS;
  float* CTX = V + OUT_ELEMS;
  float* QN  = CTX + OUT_ELEMS;          // BH*S floats
  float* KN  = QN + (size_t)BH * Sc;

  float* out  = (float*)d_out;           // [B,S,HID]
  float* attn = out + OUT_ELEMS;         // [B,NH,S,S]

  const dim3 gGemm(MTOT / 64, HIDc / 64);

  gemm_xwT_kernel<<<gGemm, 128, 0, stream>>>(query,  Wq, bq, nullptr, Q);
  gemm_xwT_kernel<<<gGemm, 128, 0, stream>>>(key_in, Wk, bk, nullptr, K);
  gemm_xwT_kernel<<<gGemm, 128, 0, stream>>>(value,  Wv, bv, nullptr, V);

  head_norms_kernel<<<(BH * Sc) / 4, 128, 0, stream>>>(Q, K, QN, KN);

  scores_kernel<<<dim3(Sc / 64, Sc / 64, BH), 128, 0, stream>>>(Q, K, QN, KN, sw, attn);

  softmax_rows_kernel<<<BH * Sc, 256, 0, stream>>>(attn);

  ctx_kernel<<<dim3(Sc / 64, Dc / 64, BH), 128, 0, stream>>>(attn, V, CTX);

  gemm_xwT_kernel<<<gGemm, 128, 0, stream>>>(CTX, Wo, bo, query, out);

  layernorm_kernel<<<MTOT, 256, 0, stream>>>(out, gamma, beta);
}